// NaiveBias2d_30451318128837
// MI455X (gfx1250) — compile-verified
//
#include <hip/hip_runtime.h>

typedef __attribute__((ext_vector_type(2))) float v2f;
typedef __attribute__((ext_vector_type(8))) float v8f;

#define SHAPE   32
#define NHEADS  8
#define SEQ     (SHAPE * SHAPE)      // 1024
#define HDIM    64
#define WLEN    (2 * SHAPE - 1)      // 63
#define LSTRIDE (NHEADS * HDIM)      // 512 floats between consecutive l

// LDS row strides (floats), padded for bank-conflict-free access
#define SV_STRIDE 68                 // 64 + 4 (keeps 16B alignment for float4)
#define B_STRIDE  33

__global__ __launch_bounds__(512)
void bias2d_wmma_kernel(const float* __restrict__ v,
                        const float* __restrict__ w,
                        float* __restrict__ out) {
    __shared__ float sE[SHAPE * SV_STRIDE];   // sE[c][d] = sum_e v[(c,e),d]
    __shared__ float sC[SHAPE * SV_STRIDE];   // sC[e][d] = sum_c v[(c,e),d]
    __shared__ float sB[SHAPE * B_STRIDE];    // bias[j][i]
    __shared__ float sT1[SHAPE * SV_STRIDE];  // T1[b][d] = bias[b][:] . sC[:][d]
    __shared__ float sT2[SHAPE * SV_STRIDE];  // T2[a][d] = bias[a][:] . sE[:][d]

    const int tid = threadIdx.x;
    const int n = blockIdx.x >> 3;
    const int h = blockIdx.x & 7;

    // ---- bias into LDS: bias[j][i] = w[h][(32 - i + j) % 63]  (uniform trip count)
    for (int idx = tid; idx < SHAPE * SHAPE; idx += 512) {
        const int j = idx >> 5, i = idx & 31;
        sB[j * B_STRIDE + i] = w[h * WLEN + ((SHAPE - i + j) % WLEN)];
    }

    // ---- reductions of v over the 32x32 (c,e) grid, float4-vectorized over d
    const int q = tid & 15;   // d-block: d = 4*q .. 4*q+3
    const int g = tid >> 4;   // 0..31 : row of the grid handled by this thread
    const float* vb = v + ((size_t)n * SEQ * LSTRIDE) + (size_t)h * HDIM + (size_t)q * 4;

    {   // pass A: sE[c][d] = sum over e (global, first touch of v)
        const int c = g;
        float4 acc = make_float4(0.f, 0.f, 0.f, 0.f);
        #pragma unroll
        for (int e = 0; e < SHAPE; ++e) {
            const float4 val =
                *reinterpret_cast<const float4*>(vb + (size_t)(c * SHAPE + e) * LSTRIDE);
            acc.x += val.x; acc.y += val.y; acc.z += val.z; acc.w += val.w;
        }
        *reinterpret_cast<float4*>(&sE[c * SV_STRIDE + q * 4]) = acc;
    }
    {   // pass B: sC[e][d] = sum over c (re-read hits L2: whole v << 192MB L2)
        const int e = g;
        float4 acc = make_float4(0.f, 0.f, 0.f, 0.f);
        #pragma unroll
        for (int c = 0; c < SHAPE; ++c) {
            const float4 val =
                *reinterpret_cast<const float4*>(vb + (size_t)(c * SHAPE + e) * LSTRIDE);
            acc.x += val.x; acc.y += val.y; acc.z += val.z; acc.w += val.w;
        }
        *reinterpret_cast<float4*>(&sC[e * SV_STRIDE + q * 4]) = acc;
    }
    __syncthreads();

    // ---- WMMA: T1 = bias x sC, T2 = bias x sE (each 32x64 = 2x4 tiles of 16x16, K=32)
    // 16 waves, one 16x16 tile each, 8 chained v_wmma_f32_16x16x4_f32 per tile.
    {
        const int wave = tid >> 5;            // 0..15
        const int lane = tid & 31;
        const int m16  = lane & 15;           // M (A/C rows) or N (B/C cols) within tile
        const int kh   = lane >> 4;           // K-half select (A/B) & row-half (C/D)
        const int useE = wave >> 3;           // 0 -> sC/T1, 1 -> sE/T2
        const int r    = wave & 7;
        const int boff = (r >> 2) << 4;       // tile row offset: 0 | 16
        const int doff = (r & 3) << 4;        // tile col offset: 0 | 16 | 32 | 48
        const float* S = useE ? sE : sC;
        float*       T = useE ? sT2 : sT1;

        v8f acc = {};
        #pragma unroll
        for (int k0 = 0; k0 < SHAPE; k0 += 4) {
            const int ka = k0 + kh * 2;       // this lane's K base within the 16x4 A frag
            v2f a, b;
            a.x = sB[(boff + m16) * B_STRIDE + ka];
            a.y = sB[(boff + m16) * B_STRIDE + ka + 1];
            b.x = S[ka * SV_STRIDE + doff + m16];
            b.y = S[(ka + 1) * SV_STRIDE + doff + m16];
            // D = A(16x4) * B(4x16) + C, exact f32
            acc = __builtin_amdgcn_wmma_f32_16x16x4_f32(
                /*neg_a=*/false, a, /*neg_b=*/false, b,
                /*c_mod=*/(short)0, acc, /*reuse_a=*/false, /*reuse_b=*/false);
        }
        // C/D layout: VGPR r8 holds rows (r8, r8+8) split across lane halves, N = lane%16
        #pragma unroll
        for (int r8 = 0; r8 < 8; ++r8) {
            T[(boff + r8 + 8 * kh) * SV_STRIDE + doff + m16] = acc[r8];
        }
    }
    __syncthreads();

    // ---- writeout: out[n, a*32+b, h, d] = T1[b][d] + T2[a][d]  (float4, coalesced)
    float* ob = out + ((size_t)n * SEQ * LSTRIDE) + (size_t)h * HDIM;
    #pragma unroll 4
    for (int it = 0; it < 32; ++it) {
        const int idx = it * 512 + tid;       // 0..16383 over (j, d/4)
        const int j  = idx >> 4;
        const int qq = idx & 15;
        const int a  = j >> 5;
        const int bb = j & 31;
        const float4 t1 = *reinterpret_cast<const float4*>(&sT1[bb * SV_STRIDE + qq * 4]);
        const float4 t2 = *reinterpret_cast<const float4*>(&sT2[a  * SV_STRIDE + qq * 4]);
        float4 o;
        o.x = t1.x + t2.x; o.y = t1.y + t2.y; o.z = t1.z + t2.z; o.w = t1.w + t2.w;
        *reinterpret_cast<float4*>(ob + (size_t)j * LSTRIDE + qq * 4) = o;
    }
}

extern "C" void kernel_launch(void* const* d_in, const int* in_sizes, int n_in,
                              void* d_out, int out_size, void* d_ws, size_t ws_size,
                              hipStream_t stream) {
    (void)in_sizes; (void)n_in; (void)out_size; (void)d_ws; (void)ws_size;
    const float* v = (const float*)d_in[0];   // (8, 1024, 8, 64) f32
    const float* w = (const float*)d_in[1];   // (1, 8, 63) f32
    float* out = (float*)d_out;               // (8, 1024, 8, 64) f32

    // one block per (n, h); 512 threads = 16 wave32 waves (one WMMA tile each)
    bias2d_wmma_kernel<<<dim3(8 * NHEADS), dim3(512), 0, stream>>>(v, w, out);
}